// test_soft_add_5935644803606
// MI455X (gfx1250) — compile-verified
//
#include <hip/hip_runtime.h>

typedef __attribute__((ext_vector_type(2))) float v2f;
typedef __attribute__((ext_vector_type(8))) float v8f;

#define NTHREADS 256
#define NBLOCKS  1024
#define TILE     256        // samples per tile == NTHREADS
#define ROW      25
#define NWAVES   (NTHREADS / 32)
#define TILE_DW  (TILE * ROW)        // 6400 dwords per array per tile
#define TILE_V4  (TILE_DW / 4)       // 1600 float4 per array per tile
#define V4_ITERS ((TILE_V4 + NTHREADS - 1) / NTHREADS)   // 7

// Kernel 1: streaming partial reduction.
// Stage 256-sample tiles of pre/tar through LDS with coalesced b128 streams
// (tile base is 16B-aligned: tile*25600 bytes), compute 6x 4-way softmax +
// squared diff per sample from LDS (stride 25 is coprime with 64 banks ->
// conflict-free), accumulate per-thread, reduce per-wave with
// V_WMMA_F32_16X16X4_F32 (+shfl), then per-block via LDS.
__global__ __launch_bounds__(NTHREADS) void soft_mse_partial(
    const float* __restrict__ pre,
    const float* __restrict__ tar,
    float* __restrict__ partials,
    int B)
{
    __shared__ __align__(16) float sPre[TILE_DW];
    __shared__ __align__(16) float sTar[TILE_DW];
    __shared__ float waveSums[NWAVES];

    const int tid = threadIdx.x;
    const long totalDw = (long)B * ROW;
    const int numTiles = (B + TILE - 1) / TILE;

    float acc = 0.0f;

    for (int tile = blockIdx.x; tile < numTiles; tile += gridDim.x) {
        const long base = (long)tile * TILE_DW;

        if ((long)(tile + 1) * TILE <= (long)B) {
            // Full tile: vectorized b128 staging.
            const float4* __restrict__ p4 = (const float4*)(pre + base);
            const float4* __restrict__ t4 = (const float4*)(tar + base);
            float4* __restrict__ sP4 = (float4*)sPre;
            float4* __restrict__ sT4 = (float4*)sTar;
            #pragma unroll
            for (int k = 0; k < V4_ITERS; ++k) {
                int i = k * NTHREADS + tid;
                if (i < TILE_V4) {
                    sP4[i] = p4[i];
                    sT4[i] = t4[i];
                }
            }
        } else {
            // Ragged last tile: guarded scalar staging.
            #pragma unroll
            for (int k = 0; k < ROW; ++k) {
                int i = k * NTHREADS + tid;
                long g = base + i;
                bool ok = (g < totalDw);
                sPre[i] = ok ? pre[g] : 0.0f;
                sTar[i] = ok ? tar[g] : 0.0f;
            }
        }
        __syncthreads();

        const int s = tile * TILE + tid;
        if (s < B) {
            const float* p = &sPre[tid * ROW];
            const float* t = &sTar[tid * ROW];
            #pragma unroll
            for (int g = 0; g < 6; ++g) {
                float x0 = p[4*g + 0], x1 = p[4*g + 1];
                float x2 = p[4*g + 2], x3 = p[4*g + 3];
                float m  = fmaxf(fmaxf(x0, x1), fmaxf(x2, x3));
                float e0 = __expf(x0 - m), e1 = __expf(x1 - m);
                float e2 = __expf(x2 - m), e3 = __expf(x3 - m);
                // 3 * EXPAND * softmax; v_rcp_f32 (~1 ulp) instead of div fixup
                float inv = 3.0f * __builtin_amdgcn_rcpf(e0 + e1 + e2 + e3);
                float d0 = fmaf(e0, inv, -t[4*g + 0]);
                float d1 = fmaf(e1, inv, -t[4*g + 1]);
                float d2 = fmaf(e2, inv, -t[4*g + 2]);
                float d3 = fmaf(e3, inv, -t[4*g + 3]);
                acc = fmaf(d0, d0, acc);
                acc = fmaf(d1, d1, acc);
                acc = fmaf(d2, d2, acc);
                acc = fmaf(d3, d3, acc);
            }
        }
        __syncthreads();
    }

    // ---- Wave32 reduction via WMMA (EXEC is all-ones here) ----
    // D = ones(16x4) * Bmat(4x16) + 0. Bmat VGPR0 = per-lane acc, VGPR1 = 0,
    // so every lane's D[.][n] = acc[n] + acc[n+16]; lane L holds column L%16.
    v2f a; a[0] = 1.0f; a[1] = 1.0f;
    v2f b; b[0] = acc;  b[1] = 0.0f;
    v8f c = {};
    c = __builtin_amdgcn_wmma_f32_16x16x4_f32(
            /*neg_a=*/false, a, /*neg_b=*/false, b,
            /*c_mod=*/(short)0, c, /*reuse_a=*/false, /*reuse_b=*/false);
    float v = c[0];
    v += __shfl_xor(v, 8, 32);
    v += __shfl_xor(v, 4, 32);
    v += __shfl_xor(v, 2, 32);
    v += __shfl_xor(v, 1, 32);

    const int wave = tid >> 5;
    const int lane = tid & 31;
    if (lane == 0) waveSums[wave] = v;
    __syncthreads();
    if (tid == 0) {
        float blockSum = 0.0f;
        #pragma unroll
        for (int w = 0; w < NWAVES; ++w) blockSum += waveSums[w];
        partials[blockIdx.x] = blockSum;
    }
}

// Kernel 2: reduce block partials in double, divide by 24*B, write scalar.
__global__ __launch_bounds__(256) void soft_mse_final(
    const float* __restrict__ partials,
    int nPartials,
    float* __restrict__ out,
    int B)
{
    __shared__ double red[256];
    double acc = 0.0;
    for (int i = threadIdx.x; i < nPartials; i += 256)
        acc += (double)partials[i];
    red[threadIdx.x] = acc;
    __syncthreads();
    for (int s = 128; s > 0; s >>= 1) {
        if (threadIdx.x < (unsigned)s) red[threadIdx.x] += red[threadIdx.x + s];
        __syncthreads();
    }
    if (threadIdx.x == 0)
        out[0] = (float)(red[0] / (24.0 * (double)B));
}

extern "C" void kernel_launch(void* const* d_in, const int* in_sizes, int n_in,
                              void* d_out, int out_size, void* d_ws, size_t ws_size,
                              hipStream_t stream)
{
    const float* pre = (const float*)d_in[0];
    const float* tar = (const float*)d_in[1];
    float* out = (float*)d_out;
    float* partials = (float*)d_ws;

    const int B = in_sizes[0] / ROW;   // 2097152 for the reference shapes

    int nb = NBLOCKS;
    long maxPartials = (long)(ws_size / sizeof(float));
    if (maxPartials < nb) nb = (maxPartials > 0) ? (int)maxPartials : 1;

    soft_mse_partial<<<nb, NTHREADS, 0, stream>>>(pre, tar, partials, B);
    soft_mse_final<<<1, 256, 0, stream>>>(partials, nb, out, B);
}